// TheirMonotonicRegressor_16810501996696
// MI455X (gfx1250) — compile-verified
//
#include <hip/hip_runtime.h>
#include <hip/hip_bf16.h>

#define BATCH 4096
#define TSEQ  200
#define FEAT  32
#define HID   8
#define FWD   5
// K layout of the fused gate GEMM: [ x_t (0..31) | z (32) | h (33..40) | pad (41..43) ]

typedef __attribute__((ext_vector_type(2))) float v2f;
typedef __attribute__((ext_vector_type(8))) float v8f;

__device__ __forceinline__ float sigm(float x) {
    return 1.0f / (1.0f + __expf(-x));
}
__device__ __forceinline__ float tanh_fast(float x) {
    // tanh(x) = 1 - 2/(e^{2x}+1); saturates correctly at +-inf
    return 1.0f - 2.0f / (1.0f + __expf(2.0f * x));
}

// Fused weight matrix Wcat[k][n]: k<33 -> W_ih[n][k], 33..40 -> W_hh[n][k-33], else 0
__device__ __forceinline__ float wcat(const float* __restrict__ W_ih,
                                      const float* __restrict__ W_hh,
                                      int k, int n) {
    if (k < FEAT + 1)       return W_ih[n * (FEAT + 1) + k];
    if (k < FEAT + 1 + HID) return W_hh[n * HID + (k - (FEAT + 1))];
    return 0.0f;
}

// prefetch one timestep of x into an A-fragment register buffer
__device__ __forceinline__ void loadx(v2f (&dst)[8], const float* __restrict__ xr, int hi) {
#pragma unroll
    for (int c = 0; c < 8; ++c)
        dst[c] = *reinterpret_cast<const v2f*>(xr + 4 * c + 2 * hi); // 8B-aligned
}

#define WMMA_F32(ACC, A, BFRAG) \
    (ACC) = __builtin_amdgcn_wmma_f32_16x16x4_f32(false, (A), false, (BFRAG), (short)0, (ACC), false, false)

__global__ __launch_bounds__(32)
void monotonic_lstm_wmma(const float* __restrict__ x,     // (B,T,F)
                         const float* __restrict__ z0,    // (1,B,1)
                         const float* __restrict__ W_ih,  // (32,33)
                         const float* __restrict__ W_hh,  // (32,8)
                         const float* __restrict__ b_ih,  // (32)
                         const float* __restrict__ b_hh,  // (32)
                         const float* __restrict__ W1,    // (5,8)
                         const float* __restrict__ b1,    // (5)
                         const float* __restrict__ W2,    // (1,5)
                         const float* __restrict__ b2,    // (1)
                         float* __restrict__ out)         // (B,T,1)
{
    __shared__ float gates_s[16 * 33];  // stride 33 -> no lane L vs L+16 bank conflict
    __shared__ float h_s[16 * 9];       // stride 9  -> de-conflict m vs m+8
    __shared__ float z_s[16];

    const int lane = threadIdx.x;   // 0..31
    const int m    = lane & 15;     // batch row within tile / N column within tile
    const int hi   = lane >> 4;     // K-half selector for A/B fragments
    const int b0   = blockIdx.x * 16;

    // ---- stage B fragments (Wcat^T) into registers: 11 K-chunks x 2 N-tiles ----
    v2f bw0[11], bw1[11];
#pragma unroll
    for (int c = 0; c < 11; ++c) {
        const int k0 = 4 * c + 2 * hi;
        bw0[c].x = wcat(W_ih, W_hh, k0,     m);
        bw0[c].y = wcat(W_ih, W_hh, k0 + 1, m);
        bw1[c].x = wcat(W_ih, W_hh, k0,     m + 16);
        bw1[c].y = wcat(W_ih, W_hh, k0 + 1, m + 16);
    }
    const float bias0 = b_ih[m]      + b_hh[m];
    const float bias1 = b_ih[m + 16] + b_hh[m + 16];

    // ---- head weights in registers ----
    float w1r[FWD * HID], b1r[FWD], w2r[FWD];
#pragma unroll
    for (int i = 0; i < FWD * HID; ++i) w1r[i] = W1[i];
#pragma unroll
    for (int i = 0; i < FWD; ++i) { b1r[i] = b1[i]; w2r[i] = W2[i]; }
    const float b2r = b2[0];

    // ---- init state: c in registers (lane owns (m, j=hi*4+u)), h/z in LDS ----
    float cst[4] = {0.f, 0.f, 0.f, 0.f};
    float zreg = z0[b0 + m];            // meaningful in lanes 0..15
    if (hi == 0) z_s[m] = zreg;
#pragma unroll
    for (int u = 0; u < 4; ++u) h_s[m * 9 + hi * 4 + u] = 0.0f;
    __builtin_amdgcn_wave_barrier();

    const float* xrow = x + ((size_t)(b0 + m) * TSEQ) * FEAT;

    // double-buffered x A-fragments: prefetch t+1 while computing t
    v2f xbuf[2][8];
    loadx(xbuf[0], xrow, hi);

#pragma unroll 2
    for (int t = 0; t < TSEQ; ++t) {
        const int cb = t & 1;
        const int tn = (t + 1 < TSEQ) ? (t + 1) : t;  // clamp: last prefetch is dead
        loadx(xbuf[cb ^ 1], xrow + (size_t)tn * FEAT, hi);

        // 4 independent accumulator chains (even/odd K-chunks x 2 gate tiles)
        v8f a0e, a0o, a1e, a1o;
#pragma unroll
        for (int r = 0; r < 8; ++r) {
            a0e[r] = bias0; a1e[r] = bias1;
            a0o[r] = 0.0f;  a1o[r] = 0.0f;
        }

        // ---- gate GEMM: 8 K-chunks from x, f32 WMMA 16x16x4 ----
#pragma unroll
        for (int c = 0; c < 8; c += 2) {
            WMMA_F32(a0e, xbuf[cb][c],     bw0[c]);
            WMMA_F32(a1e, xbuf[cb][c],     bw1[c]);
            WMMA_F32(a0o, xbuf[cb][c + 1], bw0[c + 1]);
            WMMA_F32(a1o, xbuf[cb][c + 1], bw1[c + 1]);
        }
        // ---- recurrent K-chunks 8..10 from z/h (LDS-staged previous step) ----
        {
            const float h7 = h_s[m * 9 + 7];
            v2f a;
            a.x = hi ? h_s[m * 9 + 1] : z_s[m];            // K=32 (z) / K=34 (h1)
            a.y = hi ? h_s[m * 9 + 2] : h_s[m * 9 + 0];    // K=33 (h0) / K=35 (h2)
            WMMA_F32(a0e, a, bw0[8]);
            WMMA_F32(a1e, a, bw1[8]);
            a.x = h_s[m * 9 + 3 + 2 * hi];                 // K=36 (h3) / K=38 (h5)
            a.y = h_s[m * 9 + 4 + 2 * hi];                 // K=37 (h4) / K=39 (h6)
            WMMA_F32(a0o, a, bw0[9]);
            WMMA_F32(a1o, a, bw1[9]);
            a.x = hi ? 0.0f : h7;                          // K=40 (h7) / pad (branchless)
            a.y = 0.0f;                                    // pad
            WMMA_F32(a0e, a, bw0[10]);
            WMMA_F32(a1e, a, bw1[10]);
        }
        __builtin_amdgcn_wave_barrier();

        // ---- reduce chains + scatter D tiles to LDS: gates_s[batch][gate] ----
#pragma unroll
        for (int r = 0; r < 8; ++r) {
            gates_s[(r + 8 * hi) * 33 + m]      = a0e[r] + a0o[r];
            gates_s[(r + 8 * hi) * 33 + m + 16] = a1e[r] + a1o[r];
        }
        __builtin_amdgcn_wave_barrier();

        // ---- LSTM cell: lane owns (m, j = hi*4+u), c stays in registers ----
#pragma unroll
        for (int u = 0; u < 4; ++u) {
            const int j = hi * 4 + u;
            const float ig = sigm(gates_s[m * 33 + j]);
            const float fg = sigm(gates_s[m * 33 + 8 + j]);
            const float gg = tanh_fast(gates_s[m * 33 + 16 + j]);
            const float og = sigm(gates_s[m * 33 + 24 + j]);
            const float cn = fg * cst[u] + ig * gg;
            cst[u] = cn;
            h_s[m * 9 + j] = og * tanh_fast(cn);
        }
        __builtin_amdgcn_wave_barrier();

        // ---- monotone head (lanes 0..15, one batch row each) ----
        if (hi == 0) {
            float hv[HID];
#pragma unroll
            for (int j = 0; j < HID; ++j) hv[j] = h_s[m * 9 + j];
            float delta = b2r;
#pragma unroll
            for (int k = 0; k < FWD; ++k) {
                float d = b1r[k];
#pragma unroll
                for (int j = 0; j < HID; ++j) d = fmaf(hv[j], w1r[k * HID + j], d);
                delta = fmaf(fmaxf(d, 0.0f), w2r[k], delta);
            }
            zreg += fmaxf(delta, 0.0f);
            z_s[m] = zreg;
            out[(size_t)(b0 + m) * TSEQ + t] = zreg;
        }
        __builtin_amdgcn_wave_barrier();
    }
}

extern "C" void kernel_launch(void* const* d_in, const int* in_sizes, int n_in,
                              void* d_out, int out_size, void* d_ws, size_t ws_size,
                              hipStream_t stream) {
    (void)in_sizes; (void)n_in; (void)out_size; (void)d_ws; (void)ws_size;
    const float* x    = (const float*)d_in[0];
    const float* z0   = (const float*)d_in[1];
    const float* W_ih = (const float*)d_in[2];
    const float* W_hh = (const float*)d_in[3];
    const float* b_ih = (const float*)d_in[4];
    const float* b_hh = (const float*)d_in[5];
    const float* W1   = (const float*)d_in[6];
    const float* b1   = (const float*)d_in[7];
    const float* W2   = (const float*)d_in[8];
    const float* b2   = (const float*)d_in[9];
    float* out = (float*)d_out;

    dim3 grid(BATCH / 16);   // one wave32 per 16 batch rows
    dim3 block(32);
    monotonic_lstm_wmma<<<grid, block, 0, stream>>>(x, z0, W_ih, W_hh, b_ih, b_hh,
                                                    W1, b1, W2, b2, out);
}